// GAT_GINMultiLabel_18803366822477
// MI455X (gfx1250) — compile-verified
//
#include <hip/hip_runtime.h>
#include <hip/hip_bf16.h>

typedef __attribute__((ext_vector_type(2))) float v2f;
typedef __attribute__((ext_vector_type(8))) float v8f;

#define NEG_SLOPE 0.2f
#define LN_EPS 1e-5f

// ---- order-preserving float<->uint encoding for atomicMax on floats ----
static __device__ __forceinline__ unsigned encf(float f) {
  unsigned u = __float_as_uint(f);
  return (u & 0x80000000u) ? ~u : (u | 0x80000000u);
}
static __device__ __forceinline__ float decf(unsigned u) {
  unsigned v = (u & 0x80000000u) ? (u & 0x7fffffffu) : ~u;
  return __uint_as_float(v);
}

// =====================================================================
// Generic f32 WMMA GEMM: C[M,Ncols] = act(A[M,K] x B[K,Ncols] + bias)
// One wave32 per 16x16 output tile; K stepped by 4 via V_WMMA_F32_16X16X4_F32.
// Requires M%16==0, Ncols%16==0, K%4==0 (all true for this model).
// A-frag layout (ISA 7.12.2): lanes 0-15 M=0..15, VGPR0=K+0 (lo half)/K+2 (hi),
// VGPR1=K+1/K+3.  C/D: VGPR j -> row m0+j (lo half) / m0+8+j (hi), col n0+lane%16.
// =====================================================================
template <int ACT>
__global__ __launch_bounds__(256) void wmma_gemm_f32(
    const float* __restrict__ A, const float* __restrict__ B,
    const float* __restrict__ bias, float* __restrict__ C,
    int M, int K, int Ncols) {
  const int lane = threadIdx.x & 31;
  const int wave = threadIdx.x >> 5;
  const int tile = blockIdx.x * (blockDim.x >> 5) + wave;
  const int tilesN = Ncols >> 4;
  const int tilesM = M >> 4;
  if (tile >= tilesM * tilesN) return;           // wave-uniform exit
  const int tm = tile / tilesN, tn = tile % tilesN;
  const int m0 = tm << 4, n0 = tn << 4;
  const int hf  = lane >> 4;                     // half-wave select
  const int l16 = lane & 15;
  const int arow = m0 + l16;
  const int bcol = n0 + l16;
  v8f c = {};
  for (int k = 0; k < K; k += 4) {
    const int ka = k + (hf << 1);                // k or k+2
    v2f a, b;
    a.x = A[(size_t)arow * K + ka];
    a.y = A[(size_t)arow * K + ka + 1];
    b.x = B[(size_t)(ka)     * Ncols + bcol];
    b.y = B[(size_t)(ka + 1) * Ncols + bcol];
    c = __builtin_amdgcn_wmma_f32_16x16x4_f32(false, a, false, b, (short)0, c,
                                              false, false);
  }
  const int crow0 = m0 + (hf << 3);
  const float bs = bias ? bias[bcol] : 0.0f;
#pragma unroll
  for (int j = 0; j < 8; ++j) {
    float v = c[j] + bs;
    if (ACT == 1) v = fmaxf(v, 0.0f);
    C[(size_t)(crow0 + j) * Ncols + bcol] = v;
  }
}

// repack W[h,f,c] (H=4,F,C=64) -> B[f, h*64+c]  so xh = x @ B is one GEMM
__global__ void repack_W(const float* __restrict__ W, float* __restrict__ B, int F) {
  int t = blockIdx.x * blockDim.x + threadIdx.x;
  if (t >= F * 256) return;
  int f = t >> 8, hc = t & 255, h = hc >> 6, c = hc & 63;
  B[t] = W[((size_t)(h * F + f) << 6) + c];
}

// al_s[n,h] = <xh[n,h,:], a_s[h,:]> ; al_d likewise
__global__ void attn_logits(const float* __restrict__ xh, const float* __restrict__ as_,
                            const float* __restrict__ ad_, float* __restrict__ als,
                            float* __restrict__ ald, int NN) {
  int t = blockIdx.x * blockDim.x + threadIdx.x;
  if (t >= NN * 4) return;
  int n = t >> 2, h = t & 3;
  const float* xp = xh + (size_t)n * 256 + h * 64;
  const float* sp = as_ + h * 64;
  const float* dp = ad_ + h * 64;
  float s = 0.f, d = 0.f;
#pragma unroll 8
  for (int c = 0; c < 64; ++c) { float v = xp[c]; s += v * sp[c]; d += v * dp[c]; }
  als[t] = s; ald[t] = d;
}

// pass 1: segment max of leaky_relu(al_s[s]+al_d[d]) over dst (incl self loops)
__global__ void edge_max(const float* __restrict__ als, const float* __restrict__ ald,
                         const int* __restrict__ src, const int* __restrict__ dst,
                         unsigned* __restrict__ mm, int E, int NN) {
  int t = blockIdx.x * blockDim.x + threadIdx.x;
  if (t >= (E + NN) * 4) return;
  int e = t >> 2, h = t & 3;
  int s = (e < E) ? src[e] : (e - E);
  int d = (e < E) ? dst[e] : (e - E);
  float v = als[s * 4 + h] + ald[d * 4 + h];
  v = (v > 0.f) ? v : NEG_SLOPE * v;
  atomicMax(&mm[d * 4 + h], encf(v));
}

// pass 2: den[d,h] += ex ; acc[d,h,c] += ex * xh[s,h,c]   (one wave per edge)
__global__ __launch_bounds__(256) void edge_accum(
    const float* __restrict__ als, const float* __restrict__ ald,
    const int* __restrict__ src, const int* __restrict__ dst,
    const unsigned* __restrict__ mm, float* __restrict__ den,
    float* __restrict__ acc, const float* __restrict__ xh, int E, int NN) {
  int lane = threadIdx.x & 31;
  int e = blockIdx.x * (blockDim.x >> 5) + (threadIdx.x >> 5);
  if (e >= E + NN) return;
  int s = (e < E) ? src[e] : (e - E);
  int d = (e < E) ? dst[e] : (e - E);
  float ex[4];
#pragma unroll
  for (int h = 0; h < 4; ++h) {
    float v = als[s * 4 + h] + ald[d * 4 + h];
    v = (v > 0.f) ? v : NEG_SLOPE * v;
    ex[h] = __expf(v - decf(mm[d * 4 + h]));
  }
  if (lane < 4) atomicAdd(&den[d * 4 + lane], ex[lane & 3]);
  const float* xp = xh + (size_t)s * 256;
  float* ap = acc + (size_t)d * 256;
#pragma unroll
  for (int i = 0; i < 8; ++i) {
    int idx = lane + (i << 5);
    atomicAdd(&ap[idx], ex[i >> 1] * xp[idx]);
  }
}

// xg[n,c] = relu( mean_h( acc[n,h,c]/den[n,h] ) + bg[c] )
__global__ void gat_final(const float* __restrict__ acc, const float* __restrict__ den,
                          const float* __restrict__ bg, float* __restrict__ xg, int NN) {
  int t = blockIdx.x * blockDim.x + threadIdx.x;
  if (t >= NN * 64) return;
  int n = t >> 6, c = t & 63;
  float v = 0.f;
#pragma unroll
  for (int h = 0; h < 4; ++h)
    v += acc[(size_t)n * 256 + h * 64 + c] / den[n * 4 + h];
  v = 0.25f * v + bg[c];
  xg[t] = fmaxf(v, 0.f);
}

// GIN neighbor sum: agg[d,:] += xg[s,:]  (one wave per edge, 2 floats/lane)
__global__ __launch_bounds__(256) void edge_gin_sum(
    const float* __restrict__ xg, const int* __restrict__ src,
    const int* __restrict__ dst, float* __restrict__ agg, int E) {
  int lane = threadIdx.x & 31;
  int e = blockIdx.x * (blockDim.x >> 5) + (threadIdx.x >> 5);
  if (e >= E) return;
  int s = src[e], d = dst[e];
  const float* xp = xg + (size_t)s * 64;
  float* ap = agg + (size_t)d * 64;
  atomicAdd(&ap[lane], xp[lane]);
  atomicAdd(&ap[lane + 32], xp[lane + 32]);
}

// LayerNorm over CC channels, wave32 per row, optional residual in2 and relu
template <int CC, int RELU>
__global__ __launch_bounds__(256) void layernorm_k(
    const float* __restrict__ in1, const float* __restrict__ in2,
    const float* __restrict__ w, const float* __restrict__ b,
    float* __restrict__ out, int rows) {
  const int lane = threadIdx.x & 31;
  int r = blockIdx.x * (blockDim.x >> 5) + (threadIdx.x >> 5);
  if (r >= rows) return;
  const int V = CC / 32;
  float vals[V];
  float s = 0.f;
#pragma unroll
  for (int i = 0; i < V; ++i) {
    int c = lane + 32 * i;
    float v = in1[(size_t)r * CC + c];
    if (in2) v += in2[(size_t)r * CC + c];
    vals[i] = v; s += v;
  }
#pragma unroll
  for (int off = 16; off >= 1; off >>= 1) s += __shfl_xor(s, off, 32);
  float mu = s / CC;
  float q = 0.f;
#pragma unroll
  for (int i = 0; i < V; ++i) { float d = vals[i] - mu; q += d * d; }
#pragma unroll
  for (int off = 16; off >= 1; off >>= 1) q += __shfl_xor(q, off, 32);
  float inv = rsqrtf(q / CC + LN_EPS);
#pragma unroll
  for (int i = 0; i < V; ++i) {
    int c = lane + 32 * i;
    float y = (vals[i] - mu) * inv * w[c] + b[c];
    if (RELU) y = fmaxf(y, 0.f);
    out[(size_t)r * CC + c] = y;
  }
}

// gate[n] = <h[n,:], gw2> + gb2  (wave32 reduce)
__global__ __launch_bounds__(256) void gate_final(
    const float* __restrict__ h, const float* __restrict__ gw2,
    const float* __restrict__ gb2, float* __restrict__ gate, int NN) {
  int lane = threadIdx.x & 31;
  int n = blockIdx.x * (blockDim.x >> 5) + (threadIdx.x >> 5);
  if (n >= NN) return;
  float s = h[(size_t)n * 64 + lane] * gw2[lane] +
            h[(size_t)n * 64 + lane + 32] * gw2[lane + 32];
#pragma unroll
  for (int off = 16; off >= 1; off >>= 1) s += __shfl_xor(s, off, 32);
  if (lane == 0) gate[n] = s + gb2[0];
}

__global__ void pool_max(const float* __restrict__ gate, const int* __restrict__ batch,
                         unsigned* __restrict__ mg, int NN) {
  int n = blockIdx.x * blockDim.x + threadIdx.x;
  if (n >= NN) return;
  atomicMax(&mg[batch[n]], encf(gate[n]));
}

__global__ __launch_bounds__(256) void pool_accum(
    const float* __restrict__ gate, const int* __restrict__ batch,
    const unsigned* __restrict__ mg, float* __restrict__ deng,
    float* __restrict__ numg, const float* __restrict__ x2, int NN) {
  int lane = threadIdx.x & 31;
  int n = blockIdx.x * (blockDim.x >> 5) + (threadIdx.x >> 5);
  if (n >= NN) return;
  int g = batch[n];
  float ex = __expf(gate[n] - decf(mg[g]));
  if (lane == 0) atomicAdd(&deng[g], ex);
  const float* xp = x2 + (size_t)n * 64;
  float* np = numg + (size_t)g * 64;
  atomicAdd(&np[lane], ex * xp[lane]);
  atomicAdd(&np[lane + 32], ex * xp[lane + 32]);
}

__global__ void pool_final(const float* __restrict__ numg, const float* __restrict__ deng,
                           float* __restrict__ pool, int GG) {
  int t = blockIdx.x * blockDim.x + threadIdx.x;
  if (t >= GG * 64) return;
  pool[t] = numg[t] / deng[t >> 6];
}

__global__ void out_head(const float* __restrict__ hln, const float* __restrict__ l2w,
                         const float* __restrict__ l2b, float* __restrict__ out, int GG) {
  int t = blockIdx.x * blockDim.x + threadIdx.x;
  if (t >= GG * 6) return;
  int g = t / 6, o = t - g * 6;
  float s = l2b[o];
#pragma unroll 8
  for (int j = 0; j < 128; ++j) s += hln[(size_t)g * 128 + j] * l2w[j * 6 + o];
  out[t] = s;
}

extern "C" void kernel_launch(void* const* d_in, const int* in_sizes, int n_in,
                              void* d_out, int out_size, void* d_ws, size_t ws_size,
                              hipStream_t stream) {
  const float* x    = (const float*)d_in[0];
  const int*   src  = (const int*)d_in[1];
  const int*   dst  = (const int*)d_in[2];
  const int*   batch= (const int*)d_in[3];
  const float* W1   = (const float*)d_in[4];
  const float* a1s  = (const float*)d_in[5];
  const float* a1d  = (const float*)d_in[6];
  const float* bg1  = (const float*)d_in[7];
  const float* m1w1 = (const float*)d_in[8];
  const float* m1b1 = (const float*)d_in[9];
  const float* m1w2 = (const float*)d_in[10];
  const float* m1b2 = (const float*)d_in[11];
  const float* ln1w = (const float*)d_in[12];
  const float* ln1b = (const float*)d_in[13];
  const float* W2   = (const float*)d_in[14];
  const float* a2s  = (const float*)d_in[15];
  const float* a2d  = (const float*)d_in[16];
  const float* bg2  = (const float*)d_in[17];
  const float* m2w1 = (const float*)d_in[18];
  const float* m2b1 = (const float*)d_in[19];
  const float* m2w2 = (const float*)d_in[20];
  const float* m2b2 = (const float*)d_in[21];
  const float* ln2w = (const float*)d_in[22];
  const float* ln2b = (const float*)d_in[23];
  const float* gw1  = (const float*)d_in[24];
  const float* gb1  = (const float*)d_in[25];
  const float* gw2  = (const float*)d_in[26];
  const float* gb2  = (const float*)d_in[27];
  const float* l1w  = (const float*)d_in[28];
  const float* l1b  = (const float*)d_in[29];
  const float* lnfw = (const float*)d_in[30];
  const float* lnfb = (const float*)d_in[31];
  const float* l2w  = (const float*)d_in[32];
  const float* l2b  = (const float*)d_in[33];
  (void)n_in; (void)ws_size;

  const int NN = in_sizes[0] / 28;   // 100000 (divisible by 16)
  const int EE = in_sizes[1];        // 1.6M
  const int GG = out_size / 6;       // 256

  // ---- workspace carve-up (floats) ----
  size_t off = 0;
  auto take = [&](size_t nel) -> size_t {
    size_t r = off; off += (nel + 63) & ~(size_t)63; return r;
  };
  float* ws = (float*)d_ws;
  float*    xh   = ws + take((size_t)NN * 256);
  float*    acc  = ws + take((size_t)NN * 256);
  float*    als  = ws + take((size_t)NN * 4);
  float*    ald  = ws + take((size_t)NN * 4);
  unsigned* mm   = (unsigned*)(ws + take((size_t)NN * 4));
  float*    den  = ws + take((size_t)NN * 4);
  float*    xg   = ws + take((size_t)NN * 64);
  float*    agg  = ws + take((size_t)NN * 64);
  float*    hbuf = ws + take((size_t)NN * 64);
  float*    gino = ws + take((size_t)NN * 64);
  float*    x1   = ws + take((size_t)NN * 64);
  float*    x2   = ws + take((size_t)NN * 64);
  float*    gate = ws + take((size_t)NN);
  float*    B1   = ws + take((size_t)28 * 256);
  float*    B2   = ws + take((size_t)64 * 256);
  unsigned* mg   = (unsigned*)(ws + take((size_t)GG));
  float*    deng = ws + take((size_t)GG);
  float*    numg = ws + take((size_t)GG * 64);
  float*    pool = ws + take((size_t)GG * 64);
  float*    h1   = ws + take((size_t)GG * 128);
  float*    hln  = ws + take((size_t)GG * 128);

  auto gat_gin_layer = [&](const float* xin, int F, const float* W,
                           const float* as_, const float* ad_, const float* bg,
                           const float* w1, const float* b1, const float* w2,
                           const float* b2, const float* lnw, const float* lnb,
                           float* Brep, float* xout) {
    int th = F * 256;
    repack_W<<<(th + 255) / 256, 256, 0, stream>>>(W, Brep, F);
    int tiles = (NN / 16) * (256 / 16);
    wmma_gemm_f32<0><<<(tiles + 7) / 8, 256, 0, stream>>>(xin, Brep, nullptr, xh, NN, F, 256);
    th = NN * 4;
    attn_logits<<<(th + 255) / 256, 256, 0, stream>>>(xh, as_, ad_, als, ald, NN);
    hipMemsetAsync(mm, 0, (size_t)NN * 4 * sizeof(unsigned), stream);
    hipMemsetAsync(den, 0, (size_t)NN * 4 * sizeof(float), stream);
    hipMemsetAsync(acc, 0, (size_t)NN * 256 * sizeof(float), stream);
    th = (EE + NN) * 4;
    edge_max<<<(th + 255) / 256, 256, 0, stream>>>(als, ald, src, dst, mm, EE, NN);
    int waves = EE + NN;
    edge_accum<<<(waves + 7) / 8, 256, 0, stream>>>(als, ald, src, dst, mm, den, acc, xh, EE, NN);
    th = NN * 64;
    gat_final<<<(th + 255) / 256, 256, 0, stream>>>(acc, den, bg, xg, NN);
    hipMemcpyAsync(agg, xg, (size_t)NN * 64 * sizeof(float),
                   hipMemcpyDeviceToDevice, stream);
    edge_gin_sum<<<(EE + 7) / 8, 256, 0, stream>>>(xg, src, dst, agg, EE);
    tiles = (NN / 16) * (64 / 16);
    wmma_gemm_f32<1><<<(tiles + 7) / 8, 256, 0, stream>>>(agg, w1, b1, hbuf, NN, 64, 64);
    wmma_gemm_f32<0><<<(tiles + 7) / 8, 256, 0, stream>>>(hbuf, w2, b2, gino, NN, 64, 64);
    layernorm_k<64, 0><<<(NN + 7) / 8, 256, 0, stream>>>(xg, gino, lnw, lnb, xout, NN);
  };

  gat_gin_layer(x,  28, W1, a1s, a1d, bg1, m1w1, m1b1, m1w2, m1b2, ln1w, ln1b, B1, x1);
  gat_gin_layer(x1, 64, W2, a2s, a2d, bg2, m2w1, m2b1, m2w2, m2b2, ln2w, ln2b, B2, x2);

  // ---- global attention pooling ----
  int tiles = (NN / 16) * (64 / 16);
  wmma_gemm_f32<1><<<(tiles + 7) / 8, 256, 0, stream>>>(x2, gw1, gb1, hbuf, NN, 64, 64);
  gate_final<<<(NN + 7) / 8, 256, 0, stream>>>(hbuf, gw2, gb2, gate, NN);
  hipMemsetAsync(mg, 0, (size_t)GG * sizeof(unsigned), stream);
  hipMemsetAsync(deng, 0, (size_t)GG * sizeof(float), stream);
  hipMemsetAsync(numg, 0, (size_t)GG * 64 * sizeof(float), stream);
  pool_max<<<(NN + 255) / 256, 256, 0, stream>>>(gate, batch, mg, NN);
  pool_accum<<<(NN + 7) / 8, 256, 0, stream>>>(gate, batch, mg, deng, numg, x2, NN);
  pool_final<<<(GG * 64 + 255) / 256, 256, 0, stream>>>(numg, deng, pool, GG);

  // ---- head: relu(LN(pool @ l1w + l1b)) @ l2w + l2b ----
  tiles = (GG / 16) * (128 / 16);
  wmma_gemm_f32<0><<<(tiles + 7) / 8, 256, 0, stream>>>(pool, l1w, l1b, h1, GG, 64, 128);
  layernorm_k<128, 1><<<(GG + 7) / 8, 256, 0, stream>>>(h1, nullptr, lnfw, lnfb, hln, GG);
  out_head<<<(GG * 6 + 255) / 256, 256, 0, stream>>>(hln, l2w, l2b, (float*)d_out, GG);
}